// LightGCN_74869869904145
// MI455X (gfx1250) — compile-verified
//
#include <hip/hip_runtime.h>
#include <hip/hip_bf16.h>
#include <math.h>

typedef __attribute__((ext_vector_type(2))) float v2f;
typedef __attribute__((ext_vector_type(8))) float v8f;

#define DNODE 64

// ---------------------------------------------------------------------------
// deg[dst[e]] += 1
__global__ void deg_kernel(const int* __restrict__ dst, int E, float* __restrict__ deg) {
    int e = blockIdx.x * blockDim.x + threadIdx.x;
    if (e < E) atomicAdd(&deg[dst[e]], 1.0f);
}

// dinv[v] = deg>0 ? deg^-0.5 : 0   (in place)
__global__ void dinv_kernel(float* __restrict__ deg, int n) {
    int i = blockIdx.x * blockDim.x + threadIdx.x;
    if (i < n) {
        float d = deg[i];
        deg[i] = (d > 0.0f) ? (1.0f / sqrtf(d)) : 0.0f;
    }
}

// h_out[dst] += dinv[src]*dinv[dst] * h_in[src]   (16 threads/edge, float4 each)
__global__ void prop_kernel(const float* __restrict__ hin, float* __restrict__ hout,
                            const int* __restrict__ src, const int* __restrict__ dst,
                            const float* __restrict__ dinv, int E) {
    unsigned t = blockIdx.x * blockDim.x + threadIdx.x;
    unsigned e = t >> 4;
    if (e >= (unsigned)E) return;
    int c = (int)(t & 15u) * 4;
    int s = src[e], d = dst[e];
    float w = dinv[s] * dinv[d];
    if (w != 0.0f) {
        const float4 v = *(const float4*)(hin + (size_t)s * DNODE + c);
        float* o = hout + (size_t)d * DNODE + c;
        atomicAdd(o + 0, v.x * w);
        atomicAdd(o + 1, v.y * w);
        atomicAdd(o + 2, v.z * w);
        atomicAdd(o + 3, v.w * w);
    }
}

__global__ void addsum_kernel(float* __restrict__ hsum, const float* __restrict__ a,
                              const float* __restrict__ b, int n) {
    int i = blockIdx.x * blockDim.x + threadIdx.x;
    if (i < n) hsum[i] = a[i] + b[i];
}

__global__ void accum_kernel(float* __restrict__ hsum, const float* __restrict__ x, int n) {
    int i = blockIdx.x * blockDim.x + threadIdx.x;
    if (i < n) hsum[i] += x[i];
}

// sum of squares of emb rows at users/pos/neg -> acc[0]
__global__ void reg_kernel(const float* __restrict__ emb, const int* __restrict__ users,
                           const int* __restrict__ pos, const int* __restrict__ neg,
                           int B, float* __restrict__ acc) {
    __shared__ float sdata[256];
    int t = blockIdx.x * blockDim.x + threadIdx.x;
    float s = 0.0f;
    if (t < B * DNODE) {
        int b = t >> 6, d = t & 63;
        float a0 = emb[(size_t)users[b] * DNODE + d];
        float a1 = emb[(size_t)pos[b]   * DNODE + d];
        float a2 = emb[(size_t)neg[b]   * DNODE + d];
        s = a0 * a0 + a1 * a1 + a2 * a2;
    }
    sdata[threadIdx.x] = s;
    __syncthreads();
    for (int off = 128; off > 0; off >>= 1) {
        if (threadIdx.x < off) sdata[threadIdx.x] += sdata[threadIdx.x + off];
        __syncthreads();
    }
    if (threadIdx.x == 0) atomicAdd(acc, sdata[0]);
}

// BPR score stage: one wave32 per 16 batch items.
// pos_s[m]=dot(U[m],P[m]), neg_s[m]=dot(U[m],N[m]) via V_WMMA_F32_16X16X4_F32
// (diagonal of 16x16 U*P^T / U*N^T, fp32 throughout). hsum rows are 4*all_emb,
// so each dot carries a 1/16 scale applied after the diagonal extraction.
__global__ void __launch_bounds__(32)
score_kernel(const float* __restrict__ hsum, const int* __restrict__ users,
             const int* __restrict__ pos, const int* __restrict__ neg,
             float* __restrict__ acc_soft) {
    const int lane = threadIdx.x;          // 0..31, single wave
    const int b0   = blockIdx.x * 16;
    const int r    = lane & 15;            // A: row M, B: col N (same lane striping)
    const int koff = (lane >> 4) * 2;      // lanes 16-31 cover K+2,K+3

    const float* U = hsum + (size_t)users[b0 + r] * DNODE;
    const float* P = hsum + (size_t)pos[b0 + r]   * DNODE;
    const float* Q = hsum + (size_t)neg[b0 + r]   * DNODE;

    v8f accP = {};
    v8f accN = {};
#pragma unroll
    for (int k0 = 0; k0 < DNODE; k0 += 4) {
        v2f a  = *(const v2f*)(U + k0 + koff);   // A[m][k0+koff..+1]
        v2f bp = *(const v2f*)(P + k0 + koff);   // B[k0+koff..+1][n] = P[n][k]
        v2f bn = *(const v2f*)(Q + k0 + koff);
        accP = __builtin_amdgcn_wmma_f32_16x16x4_f32(
            false, a, false, bp, (short)0, accP, false, false);
        accN = __builtin_amdgcn_wmma_f32_16x16x4_f32(
            false, a, false, bn, (short)0, accN, false, false);
    }

    // C/D layout: lane L, VGPR v -> M = v + 8*(L>>4), N = L&15.
    // Diagonal (M==N): lanes 0..7 at v=lane, lanes 24..31 at v=lane-24.
    const bool hasDiag = (lane < 8) || (lane >= 24);
    const int  v       = (lane < 8) ? lane : (lane - 24);
    float dP = 0.0f, dN = 0.0f;
#pragma unroll
    for (int j = 0; j < 8; ++j) {
        dP = (v == j) ? accP[j] : dP;
        dN = (v == j) ? accN[j] : dN;
    }
    if (hasDiag) {
        float x  = (dN - dP) * (1.0f / 16.0f);            // (neg-pos), /4 per operand
        float sp = fmaxf(x, 0.0f) + log1pf(expf(-fabsf(x)));  // stable softplus
        atomicAdd(acc_soft, sp);
    }
}

__global__ void finalize_kernel(const float* __restrict__ acc, float* __restrict__ out, int B) {
    if (blockIdx.x == 0 && threadIdx.x == 0) {
        float reg = 0.5f * acc[0] / (float)B * 1e-4f;
        float le  = acc[1] / (float)B;
        out[0] = le + reg;   // loss
        out[1] = le;         // loss_emb
        out[2] = reg;        // reg_loss
    }
}

// ---------------------------------------------------------------------------
extern "C" void kernel_launch(void* const* d_in, const int* in_sizes, int n_in,
                              void* d_out, int out_size, void* d_ws, size_t ws_size,
                              hipStream_t stream) {
    const float* emb   = (const float*)d_in[0];
    const int*   src   = (const int*)d_in[1];
    const int*   dst   = (const int*)d_in[2];
    const int*   users = (const int*)d_in[3];
    const int*   pos   = (const int*)d_in[4];
    const int*   neg   = (const int*)d_in[5];

    const int N  = in_sizes[0] / DNODE;   // 100000
    const int E  = in_sizes[1];           // 1280000
    const int B  = in_sizes[3];           // 4096
    const int ND = N * DNODE;

    // workspace carve-up (all float32)
    float* ws    = (float*)d_ws;
    size_t npad  = ((size_t)N + 255) & ~(size_t)255;   // keep hA 256B-aligned
    float* dinv  = ws;                                 // [N]
    float* hA    = ws + npad;                          // [N*64]
    float* hB    = hA + (size_t)ND;                    // [N*64]
    float* hsum  = hB + (size_t)ND;                    // [N*64]
    float* acc   = hsum + (size_t)ND;                  // [2]: reg_sumsq, softplus_sum

    const int T = 256;
    const int gE   = (E + T - 1) / T;
    const int gN   = (N + T - 1) / T;
    const int gND  = (ND + T - 1) / T;
    const int gEp  = (int)(((size_t)E * 16 + T - 1) / T);
    const int gReg = (B * DNODE + T - 1) / T;

    // --- edge norm ---
    hipMemsetAsync(dinv, 0, (size_t)N * sizeof(float), stream);
    hipMemsetAsync(acc,  0, 2 * sizeof(float), stream);
    deg_kernel<<<gE, T, 0, stream>>>(dst, E, dinv);
    dinv_kernel<<<gN, T, 0, stream>>>(dinv, N);

    // --- layer 1: emb -> hA ; hsum = emb + hA ---
    hipMemsetAsync(hA, 0, (size_t)ND * sizeof(float), stream);
    prop_kernel<<<gEp, T, 0, stream>>>(emb, hA, src, dst, dinv, E);
    addsum_kernel<<<gND, T, 0, stream>>>(hsum, emb, hA, ND);

    // --- layer 2: hA -> hB ; hsum += hB ---
    hipMemsetAsync(hB, 0, (size_t)ND * sizeof(float), stream);
    prop_kernel<<<gEp, T, 0, stream>>>(hA, hB, src, dst, dinv, E);
    accum_kernel<<<gND, T, 0, stream>>>(hsum, hB, ND);

    // --- layer 3: hB -> hA ; hsum += hA ---
    hipMemsetAsync(hA, 0, (size_t)ND * sizeof(float), stream);
    prop_kernel<<<gEp, T, 0, stream>>>(hB, hA, src, dst, dinv, E);
    accum_kernel<<<gND, T, 0, stream>>>(hsum, hA, ND);

    // --- losses ---
    reg_kernel<<<gReg, T, 0, stream>>>(emb, users, pos, neg, B, &acc[0]);
    score_kernel<<<B / 16, 32, 0, stream>>>(hsum, users, pos, neg, &acc[1]);
    finalize_kernel<<<1, 32, 0, stream>>>(acc, (float*)d_out, B);
}